// ManeuverIntentGNN_781684048711
// MI455X (gfx1250) — compile-verified
//
#include <hip/hip_runtime.h>
#include <hip/hip_bf16.h>

// ---------------------------------------------------------------------------
// ManeuverIntentGNN forward for MI455X (gfx1250, wave32).
// Dense GEMMs -> V_WMMA_F32_16X16X4_F32 (full fp32 matrix pipe), one wave per
// 16x(16*NT) tile with A-fragment reuse. Dense-softmax-with-zeros rewritten:
//   h[i] = (colsum + sum_e (exp(lrelu(e))-1)*Wh[dst]) / (N + sum_e (exp-1))
// ---------------------------------------------------------------------------

#define NN   3072
#define NB   8
#define NE   98304
#define FIN  23
#define HID  128
#define NPG  (NN / NB)   // 384 nodes per graph (batch = repeat(arange(8),384))

typedef float v2f __attribute__((ext_vector_type(2)));
typedef float v8f __attribute__((ext_vector_type(8)));

// ---------------------------------------------------------------------------
// fp32 WMMA GEMM: C[M,N] = act(A[M,K] @ W[K,N] + bias)
// One wave32 per 16x(16*NT) tile; K consumed 4/step by v_wmma_f32_16x16x4_f32.
// act: 0=none 1=relu 2=sigmoid
// Fragment layouts (ISA 7.12.2, 32-bit):
//   A 16x4 : lanes 0-15 row=l hold K=k0+0/k0+1; lanes 16-31 hold K=k0+2/k0+3
//   B 4x16 : lanes 0-15 col=l hold K=k0+0/k0+1; lanes 16-31 hold K=k0+2/k0+3
//   C 16x16: vgpr r, lanes 0-15 -> (M=r, N=lane); lanes 16-31 -> (M=r+8)
// OOB A-rows / B-cols need no masking: they only feed C entries we never
// store.  Only the K-tail is zero-masked (on the A side, which suffices).
// ---------------------------------------------------------------------------
template <int NT, bool GUARD>
__global__ __launch_bounds__(256)
void gemm_wmma(const float* __restrict__ A, const float* __restrict__ W,
               const float* __restrict__ bias, float* __restrict__ C,
               int M, int K, int N, int act)
{
    const int groupsN = (N + 16 * NT - 1) / (16 * NT);
    const int wave = (blockIdx.x * blockDim.x + threadIdx.x) >> 5;
    const int lane = threadIdx.x & 31;
    const int tm = wave / groupsN;
    const int tg = wave - tm * groupsN;
    const int m0 = tm << 4;
    const int n0 = tg * (16 * NT);
    if (m0 >= M) return;                      // wave-uniform (EXEC stays full)

    const int half = lane >> 4;
    const int l16  = lane & 15;

    int rowA = m0 + l16;
    if (GUARD) rowA = rowA < M ? rowA : (M - 1);      // clamp; no zeroing needed
    const float* __restrict__ Arow = A + (size_t)rowA * K;

    int  col[NT];
    bool colOk[NT];
    #pragma unroll
    for (int j = 0; j < NT; ++j) {
        int c = n0 + j * 16 + l16;
        colOk[j] = !GUARD || (c < N);
        col[j] = colOk[j] ? c : (N - 1);              // clamp; no zeroing needed
    }

    const v8f zero = {};
    v8f acc[NT];
    #pragma unroll
    for (int j = 0; j < NT; ++j) acc[j] = zero;

    const int kaOfs = half << 1;
    const int K4 = K & ~3;
    int k0 = 0;
    #pragma unroll 2
    for (; k0 < K4; k0 += 4) {                        // branchless hot loop
        const int ka = k0 + kaOfs;
        v2f a;
        a.x = Arow[ka];
        a.y = Arow[ka + 1];
        const float* __restrict__ Wk = W + (size_t)ka * N;
        #pragma unroll
        for (int j = 0; j < NT; ++j) {
            v2f b;
            b.x = Wk[col[j]];
            b.y = Wk[N + col[j]];
            acc[j] = __builtin_amdgcn_wmma_f32_16x16x4_f32(
                         false, a, false, b, (short)0, acc[j], false, false);
        }
    }
    if (k0 < K) {                                     // K tail: clamp + select
        const int ka = k0 + kaOfs;
        const int k1 = ka     < K ? ka     : K - 1;
        const int k2 = ka + 1 < K ? ka + 1 : K - 1;
        v2f a;
        a.x = (ka     < K) ? Arow[k1] : 0.0f;         // zeroing A side suffices
        a.y = (ka + 1 < K) ? Arow[k2] : 0.0f;
        #pragma unroll
        for (int j = 0; j < NT; ++j) {
            v2f b;
            b.x = W[(size_t)k1 * N + col[j]];
            b.y = W[(size_t)k2 * N + col[j]];
            acc[j] = __builtin_amdgcn_wmma_f32_16x16x4_f32(
                         false, a, false, b, (short)0, acc[j], false, false);
        }
    }

    #pragma unroll
    for (int j = 0; j < NT; ++j) {
        const float bv = bias ? bias[col[j]] : 0.0f;  // clamped col: safe read
        #pragma unroll
        for (int r = 0; r < 8; ++r) {
            const int row = m0 + r + (half << 3);
            float v = acc[j][r] + bv;
            if (act == 1)      v = v > 0.0f ? v : 0.0f;
            else if (act == 2) v = 1.0f / (1.0f + __expf(-v));
            if (!GUARD || (row < M && colOk[j]))
                C[(size_t)row * N + col[j]] = v;
        }
    }
}

// Per-(node,head) attention half-scores: s=Wh.a[0:F], t=Wh.a[F:2F]
__global__ void gat_scores(const float* __restrict__ Wh, const float* __restrict__ a,
                           float* __restrict__ s, float* __restrict__ t, int H, int F)
{
    const int idx = blockIdx.x * blockDim.x + threadIdx.x;
    if (idx >= NN * H) return;
    const float* w = Wh + (size_t)idx * F;
    float ss = 0.0f, tt = 0.0f;
    for (int f = 0; f < F; ++f) { ss += w[f] * a[f]; tt += w[f] * a[F + f]; }
    s[idx] = ss; t[idx] = tt;
}

// Column sum over all nodes: S[h*F+f] = sum_i Wh[i,h,f]
__global__ void gat_colsum(const float* __restrict__ Wh, float* __restrict__ S,
                           int H, int F)
{
    const int idx = blockIdx.x * blockDim.x + threadIdx.x;
    if (idx >= H * F) return;
    const int hf = H * F;
    float acc = 0.0f;
    for (int i = 0; i < NN; ++i) acc += Wh[(size_t)i * hf + idx];
    S[idx] = acc;
}

// One wave per (edge, head): w = exp(lrelu(s_src+t_dst)) - 1
// num[src,h,:] += w * Wh[dst,h,:] ; denom[src,h] += w    (F multiple of 4)
__global__ void gat_edge_agg(const long long* __restrict__ ei,
                             const float* __restrict__ s, const float* __restrict__ t,
                             const float* __restrict__ Wh,
                             float* __restrict__ num, float* __restrict__ denom,
                             int H, int F)
{
    const int wave = (blockIdx.x * blockDim.x + threadIdx.x) >> 5;
    const int lane = threadIdx.x & 31;
    if (wave >= NE * H) return;                // wave-uniform
    const int e = wave / H;
    const int h = wave - e * H;
    const int src = (int)ei[e];
    const int dst = (int)ei[NE + e];
    float sc = s[src * H + h] + t[dst * H + h];
    sc = sc > 0.0f ? sc : 0.2f * sc;           // leaky_relu(0.2)
    const float w = __expf(sc) - 1.0f;
    const float* whd = Wh  + ((size_t)dst * H + h) * F;
    float*       nm  = num + ((size_t)src * H + h) * F;
    for (int f0 = lane * 4; f0 < F; f0 += 128) {
        const float4 w4 = *(const float4*)(whd + f0);   // 16B-aligned (F%4==0)
        atomicAdd(&nm[f0 + 0], w * w4.x);
        atomicAdd(&nm[f0 + 1], w * w4.y);
        atomicAdd(&nm[f0 + 2], w * w4.z);
        atomicAdd(&nm[f0 + 3], w * w4.w);
    }
    if (lane == 0) atomicAdd(&denom[src * H + h], w);
}

// out[i,f] = relu( mean_h (S[h,f] + num[i,h,f]) / (N + denom[i,h]) )
__global__ void gat_finalize(const float* __restrict__ S, const float* __restrict__ num,
                             const float* __restrict__ denom, float* __restrict__ out,
                             int H, int F)
{
    const int idx = blockIdx.x * blockDim.x + threadIdx.x;
    if (idx >= NN * F) return;
    const int i = idx / F, f = idx - i * F;
    float acc = 0.0f;
    for (int h = 0; h < H; ++h)
        acc += (S[h * F + f] + num[((size_t)i * H + h) * F + f])
               / ((float)NN + denom[i * H + h]);
    acc *= (1.0f / (float)H);
    out[idx] = acc > 0.0f ? acc : 0.0f;
}

// Segment mean/max pooling over contiguous 384-node graphs -> g[8,128]
__global__ void pool_graphs(const float* __restrict__ h3, float* __restrict__ g)
{
    const int idx = blockIdx.x * blockDim.x + threadIdx.x;
    if (idx >= NB * 64) return;
    const int b = idx >> 6, f = idx & 63;
    const float* p = h3 + (size_t)b * NPG * 64 + f;
    float sm = 0.0f, mx = -3.0e38f;
    for (int n = 0; n < NPG; ++n) {
        const float v = p[(size_t)n * 64];
        sm += v;
        mx = v > mx ? v : mx;
    }
    g[b * 128 + f]      = sm * (1.0f / (float)NPG);
    g[b * 128 + 64 + f] = mx;
}

// ---------------------------------------------------------------------------
static inline void launch_gemm(const float* A, const float* W, const float* bias,
                               float* C, int M, int K, int N, int act,
                               hipStream_t stream)
{
    if (M % 16 == 0 && N % 64 == 0) {
        const long tiles = (long)(M / 16) * (N / 64);
        gemm_wmma<4, false><<<(int)((tiles + 7) / 8), 256, 0, stream>>>(
            A, W, bias, C, M, K, N, act);
    } else if (M % 16 == 0 && N % 32 == 0) {
        const long tiles = (long)(M / 16) * (N / 32);
        gemm_wmma<2, false><<<(int)((tiles + 7) / 8), 256, 0, stream>>>(
            A, W, bias, C, M, K, N, act);
    } else {
        const long tiles = (long)((M + 15) / 16) * ((N + 15) / 16);
        gemm_wmma<1, true><<<(int)((tiles + 7) / 8), 256, 0, stream>>>(
            A, W, bias, C, M, K, N, act);
    }
}

extern "C" void kernel_launch(void* const* d_in, const int* in_sizes, int n_in,
                              void* d_out, int out_size, void* d_ws, size_t ws_size,
                              hipStream_t stream)
{
    const float*     x     = (const float*)d_in[0];
    const long long* ei    = (const long long*)d_in[1];   // int64 [2,E]
    const float *ne_w1 = (const float*)d_in[3],  *ne_b1 = (const float*)d_in[4];
    const float *ne_w2 = (const float*)d_in[5],  *ne_b2 = (const float*)d_in[6];
    const float *g1_W  = (const float*)d_in[7],  *g1_a  = (const float*)d_in[8];
    const float *g2_W  = (const float*)d_in[9],  *g2_a  = (const float*)d_in[10];
    const float *g3_W  = (const float*)d_in[11], *g3_a  = (const float*)d_in[12];
    const float *gc_w1 = (const float*)d_in[13], *gc_b1 = (const float*)d_in[14];
    const float *gc_w2 = (const float*)d_in[15], *gc_b2 = (const float*)d_in[16];
    const float *gc_w3 = (const float*)d_in[17], *gc_b3 = (const float*)d_in[18];
    const float *nc_w1 = (const float*)d_in[19], *nc_b1 = (const float*)d_in[20];
    const float *nc_w2 = (const float*)d_in[21], *nc_b2 = (const float*)d_in[22];
    const float *cf_w1 = (const float*)d_in[23], *cf_b1 = (const float*)d_in[24];
    const float *cf_w2 = (const float*)d_in[25], *cf_b2 = (const float*)d_in[26];
    float* out = (float*)d_out;

    // ---- workspace carve (floats) ----
    float* ws   = (float*)d_ws;
    float* bufA = ws;                       // N*128  ping
    float* bufB = bufA + NN * HID;          // N*128  pong
    float* Wh   = bufB + NN * HID;          // N*1024 max
    float* num  = Wh   + NN * 1024;         // N*1024 max
    float* sbuf = num  + NN * 1024;         // N*8 max
    float* tbuf = sbuf + NN * 8;            // N*8
    float* den  = tbuf + NN * 8;            // N*8
    float* Scol = den  + NN * 8;            // 1024
    float* h3   = Scol + 1024;              // N*64
    float* g    = h3   + NN * 64;           // 8*128
    float* mlp1 = g    + NB * 128;          // max(N*32, 8*128)
    float* mlp2 = mlp1 + NN * 32;           // 8*64

    // ---- node encoder: h0 = relu(x@W1+b1)@W2+b2 ----
    launch_gemm(x,    ne_w1, ne_b1, bufA, NN, FIN, HID, 1, stream);
    launch_gemm(bufA, ne_w2, ne_b2, bufB, NN, HID, HID, 0, stream);

    // ---- three GAT layers (sparse softmax-with-zeros identity) ----
    const float* layerW[3] = {g1_W, g2_W, g3_W};
    const float* layerA[3] = {g1_a, g2_a, g3_a};
    const int    Hh[3] = {8, 4, 1};
    const int    Ff[3] = {128, 128, 64};
    const float* hin = bufB;
    float*       hout;
    for (int L = 0; L < 3; ++L) {
        const int H = Hh[L], F = Ff[L], HF = H * F;
        hout = (L == 0) ? bufA : (L == 1) ? bufB : h3;
        launch_gemm(hin, layerW[L], nullptr, Wh, NN, HID, HF, 0, stream);   // Wh
        {
            int n = NN * H;
            gat_scores<<<(n + 255) / 256, 256, 0, stream>>>(Wh, layerA[L], sbuf, tbuf, H, F);
        }
        gat_colsum<<<(HF + 255) / 256, 256, 0, stream>>>(Wh, Scol, H, F);
        hipMemsetAsync(num, 0, (size_t)NN * HF * sizeof(float), stream);
        hipMemsetAsync(den, 0, (size_t)NN * H  * sizeof(float), stream);
        {
            long waves = (long)NE * H;
            int blocks = (int)((waves + 7) / 8);
            gat_edge_agg<<<blocks, 256, 0, stream>>>(ei, sbuf, tbuf, Wh, num, den, H, F);
        }
        {
            int n = NN * F;
            gat_finalize<<<(n + 255) / 256, 256, 0, stream>>>(Scol, num, den, hout, H, F);
        }
        hin = hout;
    }

    // ---- pooling: g = [segment_mean | segment_max]  [8,128] ----
    pool_graphs<<<(NB * 64 + 255) / 256, 256, 0, stream>>>(h3, g);

    // ---- graph classifier -> out[0:32) ----
    launch_gemm(g,    gc_w1, gc_b1, mlp1, NB, 128, HID, 1, stream);
    launch_gemm(mlp1, gc_w2, gc_b2, mlp2, NB, HID, 64,  1, stream);
    launch_gemm(mlp2, gc_w3, gc_b3, out,  NB, 64,  4,   0, stream);

    // ---- node classifier -> out[32 : 32+12288) ----
    launch_gemm(h3,   nc_w1, nc_b1, mlp1,     NN, 64, 32, 1, stream);
    launch_gemm(mlp1, nc_w2, nc_b2, out + 32, NN, 32, 4,  0, stream);

    // ---- confidence head -> out[32+12288 : 32+12288+8) ----
    launch_gemm(g,    cf_w1, cf_b1, mlp1,                 NB, 128, 32, 1, stream);
    launch_gemm(mlp1, cf_w2, cf_b2, out + 32 + NN * 4,    NB, 32,  1,  2, stream);

    (void)in_sizes; (void)n_in; (void)out_size; (void)ws_size;
}